// Model_12421045420583
// MI455X (gfx1250) — compile-verified
//
#include <hip/hip_runtime.h>
#include <math.h>

// ---------------- problem constants ----------------
#define TLEN 8192
#define FLEN 4097           // TLEN/2 + 1
#define NB   32
#define NC   128
#define NSPLIT 8
#define KMODES 5
#define VMD_THREADS 512
#define VMD_SLOTS   9       // 512*8 = 4096, slot 8 covers f = 4096 (tid 0 only)
#define DFT_WAVES 4

// ---------------- ws layout (float offsets) ----------------
#define WS_PSUM   0                     // 32*8*128 = 32768
#define WS_PSQ    32768                 // 32768
#define WS_CHVAR  65536                 // 32
#define WS_REP    65568                 // 1 int (aliased)
#define WS_SIG    65600                 // 8192
#define WS_SHRE   73792                 // 4097
#define WS_SHIM   77889                 // 4097
// total 81986 floats = ~328 KB

typedef __attribute__((ext_vector_type(2))) float v2f;
typedef __attribute__((ext_vector_type(8))) float v8f;

// ============================================================
// Stage 1a: partial sums for per-(b,c) variance.
// grid = 32*8 blocks, 128 threads (thread = channel -> coalesced rows).
__global__ __launch_bounds__(128) void k_var_partial(const float* __restrict__ x,
                                                     float* __restrict__ psum,
                                                     float* __restrict__ psq) {
    int b = blockIdx.x >> 3;
    int s = blockIdx.x & 7;
    int c = threadIdx.x;
    const int tchunk = TLEN / NSPLIT;   // 1024
    const float* p = x + ((size_t)b * TLEN + (size_t)s * tchunk) * NC + c;
    double sm = 0.0, sq = 0.0;
    for (int t = 0; t < tchunk; ++t) {
        float v = p[(size_t)t * NC];
        sm += (double)v;
        sq += (double)v * (double)v;
    }
    psum[(b * NSPLIT + s) * NC + c] = (float)sm;
    psq [(b * NSPLIT + s) * NC + c] = (float)sq;
}

// Stage 1b: combine partials -> per-b summed channel variance (argmax-invariant).
__global__ __launch_bounds__(128) void k_var_combine(const float* __restrict__ psum,
                                                     const float* __restrict__ psq,
                                                     float* __restrict__ chvar) {
    __shared__ float red[128];
    int b = blockIdx.x, c = threadIdx.x;
    double S = 0.0, Q = 0.0;
    for (int s = 0; s < NSPLIT; ++s) {
        S += (double)psum[(b * NSPLIT + s) * NC + c];
        Q += (double)psq [(b * NSPLIT + s) * NC + c];
    }
    double var = (Q - S * S / (double)TLEN) / (double)(TLEN - 1);  // ddof=1
    red[c] = (float)var;
    __syncthreads();
    for (int st = 64; st > 0; st >>= 1) {
        if (c < st) red[c] += red[c + st];
        __syncthreads();
    }
    if (c == 0) chvar[b] = red[0];
}

// Stage 1c: argmax over 32 batch entries (first max wins, like jnp.argmax).
__global__ void k_argmax(const float* __restrict__ chvar, int* __restrict__ rep) {
    if (threadIdx.x == 0) {
        float bv = chvar[0];
        int bi = 0;
        for (int i = 1; i < NB; ++i)
            if (chvar[i] > bv) { bv = chvar[i]; bi = i; }
        *rep = bi;
    }
}

// Stage 1d: gather x[rep, :, 0] into contiguous sig[8192].
__global__ __launch_bounds__(256) void k_gather(const float* __restrict__ x,
                                                const int* __restrict__ repp,
                                                float* __restrict__ sig) {
    int t = blockIdx.x * 256 + threadIdx.x;
    int rep = *repp;
    if (t < TLEN) sig[t] = x[((size_t)rep * TLEN + t) * NC];
}

// ============================================================
// Stage 2: rfft(sig) via WMMA DFT matmul (V_WMMA_F32_16X16X4_F32).
// 4 waves per 16-frequency tile, each wave takes every 4th K-chunk.
// A = cos/sin basis read from an LDS sine table indexed by the exact
// integer phase p = (f*t) mod 8192 (cos = tab[(p+2048)&8191]); the phase
// advances incrementally (p += 16f mod 8192) so the hot loop has no trig
// and no integer multiply. B = 4 signal samples replicated across all 16
// columns (column-constant B fragment per the ISA B layout).
__global__ __launch_bounds__(32 * DFT_WAVES) void k_dft(const float* __restrict__ sig,
                                                        float* __restrict__ shre,
                                                        float* __restrict__ shim) {
    __shared__ float s_sig[TLEN];
    __shared__ float s_tab[TLEN];
    __shared__ float s_pre[DFT_WAVES * 16];
    __shared__ float s_pim[DFT_WAVES * 16];

    int tid = threadIdx.x;
    const float w = 6.28318530717958647692f / 8192.0f;
    for (int i = tid; i < TLEN; i += 32 * DFT_WAVES) {
        s_sig[i] = sig[i];
        s_tab[i] = sinf((float)i * w);
    }
    __syncthreads();

    int lane = tid & 31, wv = tid >> 5;
    int fbase = blockIdx.x * 16;
    int m  = lane & 15;            // row (frequency within tile)
    int kA = (lane >> 4) * 2;      // K index of A/B element 0 for this lane half
    int f  = fbase + m;

    unsigned p0   = (unsigned)(f * (4 * wv + kA))     & 8191u;
    unsigned p1   = (unsigned)(f * (4 * wv + kA + 1)) & 8191u;
    unsigned step = (unsigned)(f * 4 * DFT_WAVES)     & 8191u;

    v8f accc = {0.f, 0.f, 0.f, 0.f, 0.f, 0.f, 0.f, 0.f};  // sum x*cos
    v8f accs = {0.f, 0.f, 0.f, 0.f, 0.f, 0.f, 0.f, 0.f};  // sum x*sin

    for (int kc = wv; kc < TLEN / 4; kc += DFT_WAVES) {
        int t0 = kc * 4;
        v2f ac, as, bv;
        ac.x = s_tab[(p0 + 2048u) & 8191u];
        ac.y = s_tab[(p1 + 2048u) & 8191u];
        as.x = s_tab[p0];
        as.y = s_tab[p1];
        bv.x = s_sig[t0 + kA];
        bv.y = s_sig[t0 + kA + 1];
        accc = __builtin_amdgcn_wmma_f32_16x16x4_f32(false, ac, false, bv,
                                                     (short)0, accc, false, false);
        accs = __builtin_amdgcn_wmma_f32_16x16x4_f32(false, as, false, bv,
                                                     (short)0, accs, false, false);
        p0 = (p0 + step) & 8191u;
        p1 = (p1 + step) & 8191u;
    }

    // All C columns identical; column 0 lives in lane 0 (rows 0-7) and
    // lane 16 (rows 8-15) per the 16x16 f32 C layout. Combine wave partials
    // in a fixed order (deterministic).
    if (lane == 0 || lane == 16) {
        int rbase = (lane == 0) ? 0 : 8;
        for (int r = 0; r < 8; ++r) {
            s_pre[wv * 16 + rbase + r] = accc[r];
            s_pim[wv * 16 + rbase + r] = accs[r];
        }
    }
    __syncthreads();
    if (tid < 16) {
        int fo = fbase + tid;
        if (fo < FLEN) {
            float re = 0.f, im = 0.f;
            for (int w2 = 0; w2 < DFT_WAVES; ++w2) {
                re += s_pre[w2 * 16 + tid];
                im += s_pim[w2 * 16 + tid];
            }
            shre[fo] = re;
            shim[fo] = -im;          // X(f) = sum x cos - i sum x sin
        }
    }
}

// ============================================================
// Stage 3: VMD fixed-point loop + spectral peak pick + top-k, single block.
__device__ __forceinline__ float block_sum(float v, volatile float* w, int tid) {
    #pragma unroll
    for (int o = 16; o > 0; o >>= 1) v += __shfl_down(v, o, 32);
    int lane = tid & 31, wid = tid >> 5;
    __syncthreads();
    if (lane == 0) ((volatile float*)w)[wid] = v;
    __syncthreads();
    if (tid == 0) {
        float r = 0.f;
        for (int i = 0; i < VMD_THREADS / 32; ++i) r += w[i];
        ((volatile float*)w)[0] = r;
    }
    __syncthreads();
    float r = w[0];
    __syncthreads();
    return r;
}

__global__ __launch_bounds__(VMD_THREADS) void k_vmd(const float* __restrict__ shre,
                                                     const float* __restrict__ shim,
                                                     const int* __restrict__ kptr,
                                                     float* __restrict__ out) {
    __shared__ float s_red[VMD_THREADS];
    __shared__ int   s_idx[VMD_THREADS];
    __shared__ float s_om[KMODES];
    __shared__ float s_bestv[KMODES];
    __shared__ int   s_besti[KMODES];

    const int tid = threadIdx.x;
    const float invT = 1.0f / 8192.0f;

    // per-thread slice of spectral state, in registers
    float uhr[KMODES][VMD_SLOTS], uhi[KMODES][VMD_SLOTS];
    float lamr[VMD_SLOTS], lami[VMD_SLOTS];
    #pragma unroll
    for (int k = 0; k < KMODES; ++k)
        #pragma unroll
        for (int j = 0; j < VMD_SLOTS; ++j) { uhr[k][j] = 0.f; uhi[k][j] = 0.f; }
    #pragma unroll
    for (int j = 0; j < VMD_SLOTS; ++j) { lamr[j] = 0.f; lami[j] = 0.f; }
    if (tid < KMODES) s_om[tid] = 0.f;
    __syncthreads();

    for (int it = 0; it < 50; ++it) {
        float om[KMODES];
        #pragma unroll
        for (int k = 0; k < KMODES; ++k) om[k] = s_om[k];   // OLD omega
        __syncthreads();

        float onum[KMODES], oden[KMODES], convacc = 0.f;
        #pragma unroll
        for (int k = 0; k < KMODES; ++k) { onum[k] = 0.f; oden[k] = 0.f; }

        #pragma unroll
        for (int j = 0; j < VMD_SLOTS; ++j) {
            int f = tid + j * VMD_THREADS;
            if (f < FLEN) {
                float sr = shre[f], si = shim[f];
                float fr = (float)f * invT;
                float sumr = 0.f, sumi = 0.f;
                #pragma unroll
                for (int k = 0; k < KMODES; ++k) { sumr += uhr[k][j]; sumi += uhi[k][j]; }
                float cr = sr - sumr + 0.5f * lamr[j];
                float ci = si - sumi + 0.5f * lami[j];
                float nsumr = 0.f, nsumi = 0.f;
                #pragma unroll
                for (int k = 0; k < KMODES; ++k) {
                    float nr = cr + uhr[k][j], ni = ci + uhi[k][j];
                    float d  = fr - om[k];
                    float den = 1.0f + 1000.0f * d * d;
                    float ur = nr / den, ui = ni / den;
                    float dr = ur - uhr[k][j], di = ui - uhi[k][j];
                    convacc += dr * dr + di * di;
                    float p = ur * ur + ui * ui;
                    onum[k] += fr * p;
                    oden[k] += p;
                    uhr[k][j] = ur; uhi[k][j] = ui;
                    nsumr += ur; nsumi += ui;
                }
                lamr[j] += 1e-6f * (sr - nsumr);   // TAU
                lami[j] += 1e-6f * (si - nsumi);
            }
        }

        for (int k = 0; k < KMODES; ++k) {
            float on = block_sum(onum[k], s_red, tid);
            float od = block_sum(oden[k], s_red, tid);
            if (tid == 0) s_om[k] = on / (od + 1e-10f);   // batch mean, B==1
        }
        float conv = block_sum(convacc, s_red, tid) * (1.0f / (5.0f * 4097.0f));
        __syncthreads();
        if (it > 0 && conv < 1e-7f) break;                // TOL; iter 0 conv = inf
    }
    __syncthreads();

    // magnitudes: |u_hat| interior, |Re| at Nyquist, 0 at DC. argmax per mode.
    for (int k = 0; k < KMODES; ++k) {
        float bv = -1.f; int bi = 0;
        #pragma unroll
        for (int j = 0; j < VMD_SLOTS; ++j) {
            int f = tid + j * VMD_THREADS;
            if (f < FLEN) {
                float mgn;
                if (f == 0)             mgn = 0.f;
                else if (f == FLEN - 1) mgn = fabsf(uhr[k][j]);
                else                    mgn = sqrtf(uhr[k][j] * uhr[k][j] + uhi[k][j] * uhi[k][j]);
                if (mgn > bv) { bv = mgn; bi = f; }
            }
        }
        s_red[tid] = bv; s_idx[tid] = bi;
        __syncthreads();
        for (int st = VMD_THREADS / 2; st > 0; st >>= 1) {
            if (tid < st) {
                float ov = s_red[tid + st]; int oi = s_idx[tid + st];
                if (ov > s_red[tid] || (ov == s_red[tid] && oi < s_idx[tid])) {
                    s_red[tid] = ov; s_idx[tid] = oi;
                }
            }
            __syncthreads();
        }
        if (tid == 0) { s_bestv[k] = s_red[0]; s_besti[k] = s_idx[0]; }
        __syncthreads();
    }

    if (tid == 0) {
        int kk = *kptr; if (kk > KMODES) kk = KMODES; if (kk < 1) kk = 1;
        float e[KMODES]; int per[KMODES]; bool used[KMODES];
        for (int k = 0; k < KMODES; ++k) {
            e[k] = s_bestv[k];
            int id = s_besti[k]; if (id < 1) id = 1;
            per[k] = TLEN / id;
            used[k] = false;
        }
        // stable top-k (descending, earlier index wins ties) = lax.top_k
        for (int i = 0; i < kk; ++i) {
            int best = 0; float bv = -1e30f;
            for (int k = 0; k < KMODES; ++k)
                if (!used[k] && e[k] > bv) { bv = e[k]; best = k; }
            used[best] = true;
            out[i] = (float)per[best];                     // sel_periods
            for (int b = 0; b < NB; ++b)                   // period_weights [B,kk]
                out[kk + b * kk + i] = bv;
        }
    }
}

// ============================================================
extern "C" void kernel_launch(void* const* d_in, const int* in_sizes, int n_in,
                              void* d_out, int out_size, void* d_ws, size_t ws_size,
                              hipStream_t stream) {
    const float* x    = (const float*)d_in[0];   // [32, 8192, 128] f32
    const int*   kptr = (const int*)  d_in[1];   // scalar k
    float* out = (float*)d_out;

    float* W     = (float*)d_ws;
    float* psum  = W + WS_PSUM;
    float* psq   = W + WS_PSQ;
    float* chvar = W + WS_CHVAR;
    int*   rep   = (int*)(W + WS_REP);
    float* sig   = W + WS_SIG;
    float* shre  = W + WS_SHRE;
    float* shim  = W + WS_SHIM;

    k_var_partial<<<NB * NSPLIT, NC, 0, stream>>>(x, psum, psq);
    k_var_combine<<<NB, NC, 0, stream>>>(psum, psq, chvar);
    k_argmax<<<1, 32, 0, stream>>>(chvar, rep);
    k_gather<<<TLEN / 256, 256, 0, stream>>>(x, rep, sig);
    k_dft<<<(FLEN + 15) / 16, 32 * DFT_WAVES, 0, stream>>>(sig, shre, shim);
    k_vmd<<<1, VMD_THREADS, 0, stream>>>(shre, shim, kptr, out);
}